// DenTargetTransformerConv_28063316312098
// MI455X (gfx1250) — compile-verified
//
#include <hip/hip_runtime.h>
#include <hip/hip_bf16.h>
#include <math.h>

typedef __attribute__((ext_vector_type(2))) float v2f;
typedef __attribute__((ext_vector_type(8))) float v8f;

// ---- monotone float<->uint encoding for atomicMax on signed floats ----
__device__ __forceinline__ unsigned fkey(float f) {
    unsigned u = __float_as_uint(f);
    return (u & 0x80000000u) ? ~u : (u | 0x80000000u);
}
__device__ __forceinline__ float fdec(unsigned k) {
    unsigned u = (k & 0x80000000u) ? (k & 0x7FFFFFFFu) : ~k;
    return __uint_as_float(u);
}

// =====================================================================
// Kernel 1: k_dst = feat@Wk + bk ; skip = feat@Wskip + bskip
// fp32 WMMA 16x16x4, one 16-row tile per block, 8 waves cover
// 4 col-tiles x 2 weight matrices. K = 64 -> 16 chained WMMAs per wave.
// A layout (16x4 f32): lanes 0-15 rows M, v0/v1 = K {0,1}; lanes 16-31 K {2,3}
// B layout (4x16 f32): lanes 0-15 cols N, v0/v1 = K {0,1}; lanes 16-31 K {2,3}
// C/D (16x16 f32): VGPR r -> M = r (lanes 0-15) / 8+r (lanes 16-31), N = lane%16
// =====================================================================
__global__ __launch_bounds__(256) void proj_wmma_kernel(
    const float* __restrict__ feat,
    const float* __restrict__ Wk,   const float* __restrict__ bk,
    const float* __restrict__ Wsk,  const float* __restrict__ bsk,
    float* __restrict__ kdst, float* __restrict__ skipb, int n)
{
    const int wave = threadIdx.x >> 5;   // 0..7
    const int lane = threadIdx.x & 31;
    const int mat  = wave >> 2;          // 0 -> Wk, 1 -> Wskip
    const int ct   = wave & 3;           // column tile (16 cols each)

    const float* W    = mat ? Wsk : Wk;
    const float* bias = mat ? bsk : bk;
    float*       out  = mat ? skipb : kdst;

    const int rowBase = blockIdx.x * 16;
    const int mrow    = rowBase + (lane & 15);
    const int arow    = (mrow < n) ? mrow : (n - 1);   // clamp loads in tail tile
    const int khalf   = (lane >> 4) << 1;              // 0 or 2
    const int ncol    = ct * 16 + (lane & 15);

    v8f acc = {};
#pragma unroll
    for (int k0 = 0; k0 < 64; k0 += 4) {
        v2f a, b;
        a.x = feat[(size_t)arow * 64 + k0 + khalf + 0];
        a.y = feat[(size_t)arow * 64 + k0 + khalf + 1];
        b.x = W[(size_t)(k0 + khalf + 0) * 64 + ncol];
        b.y = W[(size_t)(k0 + khalf + 1) * 64 + ncol];
        acc = __builtin_amdgcn_wmma_f32_16x16x4_f32(
            /*neg_a=*/false, a, /*neg_b=*/false, b,
            /*c_mod=*/(short)0, acc, /*reuse_a=*/false, /*reuse_b=*/false);
    }

    const float bv = bias[ncol];
    const int rowOff = rowBase + ((lane >> 4) << 3);   // +0 or +8
#pragma unroll
    for (int r = 0; r < 8; ++r) {
        int row = rowOff + r;
        if (row < n) out[(size_t)row * 64 + ncol] = acc[r] + bv;
    }
}

// =====================================================================
// Kernel 2: per-(edge,head) attention score + segment max (encoded atomicMax)
// =====================================================================
__global__ __launch_bounds__(256) void score_max_kernel(
    const float* __restrict__ q, const float* __restrict__ kdst,
    const int* __restrict__ src, const int* __restrict__ dst,
    float* __restrict__ ascore, unsigned* __restrict__ amax, int E)
{
    int tid = blockIdx.x * blockDim.x + threadIdx.x;
    if (tid >= E * 4) return;
    int e = tid >> 2, h = tid & 3;
    int s = src[e], d = dst[e];
    const float4* qp = (const float4*)(q    + (size_t)s * 64 + h * 16);
    const float4* kp = (const float4*)(kdst + (size_t)d * 64 + h * 16);
    float acc = 0.0f;
#pragma unroll
    for (int i = 0; i < 4; ++i) {
        float4 qa = qp[i], ka = kp[i];
        acc += qa.x * ka.x + qa.y * ka.y + qa.z * ka.z + qa.w * ka.w;
    }
    acc *= 0.25f;  // 1/sqrt(D), D=16
    ascore[tid] = acc;
    atomicMax(amax + (size_t)d * 4 + h, fkey(acc));
}

// =====================================================================
// Kernel 3: ex = exp(a - amax[dst]); denom += ex; agg += v_src * ex
// thread per (edge, channel) ; channel c in [0,64), head = c/16
// =====================================================================
__global__ __launch_bounds__(256) void agg_kernel(
    const float* __restrict__ v,
    const int* __restrict__ src, const int* __restrict__ dst,
    const float* __restrict__ ascore, const unsigned* __restrict__ amax,
    float* __restrict__ denom, float* __restrict__ agg, int E)
{
    int tid = blockIdx.x * blockDim.x + threadIdx.x;
    if (tid >= E * 64) return;
    int e = tid >> 6, c = tid & 63, h = c >> 4;
    int s = src[e], d = dst[e];
    float a  = ascore[(size_t)e * 4 + h];
    float m  = fdec(amax[(size_t)d * 4 + h]);
    float ex = __expf(a - m);
    if ((c & 15) == 0) atomicAdd(denom + (size_t)d * 4 + h, ex);
    atomicAdd(agg + (size_t)d * 64 + c, v[(size_t)s * 64 + c] * ex);
}

// =====================================================================
// Kernel 4: per-node epilogue: normalize, gate, blend, LayerNorm, PReLU
// one wave32 per node; lane owns columns (lane, lane+32)
// =====================================================================
__global__ __launch_bounds__(256) void finish_kernel(
    const float* __restrict__ skipb, const float* __restrict__ agg,
    const float* __restrict__ denom,
    const float* __restrict__ Wg, const float* __restrict__ bg,
    const float* __restrict__ gamma, const float* __restrict__ beta,
    const float* __restrict__ pa, float* __restrict__ out, int n)
{
    const int wave = threadIdx.x >> 5;
    const int lane = threadIdx.x & 31;
    const int node = blockIdx.x * 8 + wave;
    if (node >= n) return;

    const int c0 = lane, c1 = lane + 32;
    const size_t base = (size_t)node * 64;
    float sk0 = skipb[base + c0], sk1 = skipb[base + c1];
    float r0  = agg[base + c0] / denom[(size_t)node * 4 + (c0 >> 4)];
    float r1  = agg[base + c1] / denom[(size_t)node * 4 + (c1 >> 4)];

    // gate logit = [skip, rst, skip-rst] . Wgate  (192-wide)
    float part = sk0 * Wg[c0] + r0 * Wg[64 + c0] + (sk0 - r0) * Wg[128 + c0]
               + sk1 * Wg[c1] + r1 * Wg[64 + c1] + (sk1 - r1) * Wg[128 + c1];
#pragma unroll
    for (int off = 16; off > 0; off >>= 1) part += __shfl_xor(part, off, 32);
    float g = 1.0f / (1.0f + __expf(-(part + bg[0])));

    r0 = g * sk0 + (1.0f - g) * r0;
    r1 = g * sk1 + (1.0f - g) * r1;

    // LayerNorm over 64
    float s = r0 + r1;
#pragma unroll
    for (int off = 16; off > 0; off >>= 1) s += __shfl_xor(s, off, 32);
    float mu = s * (1.0f / 64.0f);
    float d0 = r0 - mu, d1 = r1 - mu;
    float vv = d0 * d0 + d1 * d1;
#pragma unroll
    for (int off = 16; off > 0; off >>= 1) vv += __shfl_xor(vv, off, 32);
    float inv = rsqrtf(vv * (1.0f / 64.0f) + 1e-5f);

    float ap = pa[0];
    float y0 = d0 * inv * gamma[c0] + beta[c0];
    float y1 = d1 * inv * gamma[c1] + beta[c1];
    out[base + c0] = (y0 >= 0.0f) ? y0 : ap * y0;
    out[base + c1] = (y1 >= 0.0f) ? y1 : ap * y1;
}

// =====================================================================
extern "C" void kernel_launch(void* const* d_in, const int* in_sizes, int n_in,
                              void* d_out, int out_size, void* d_ws, size_t ws_size,
                              hipStream_t stream)
{
    const float* q     = (const float*)d_in[0];
    const float* v     = (const float*)d_in[1];
    const float* feat  = (const float*)d_in[2];
    const int*   src   = (const int*)  d_in[3];
    const int*   dst   = (const int*)  d_in[4];
    const float* Wk    = (const float*)d_in[5];
    const float* bk    = (const float*)d_in[6];
    const float* Wsk   = (const float*)d_in[7];
    const float* bsk   = (const float*)d_in[8];
    const float* Wg    = (const float*)d_in[9];
    const float* bg    = (const float*)d_in[10];
    const float* gamma = (const float*)d_in[11];
    const float* beta  = (const float*)d_in[12];
    const float* pa    = (const float*)d_in[13];

    const int N = in_sizes[2] / 64;   // nodes (feat is [N,64])
    const int E = in_sizes[3];        // edges

    // workspace carve: kdst[N*64] | skip[N*64] | ascore[E*4] | amax[N*4] | denom[N*4] | agg[N*64]
    float*    kdst   = (float*)d_ws;
    float*    skipb  = kdst  + (size_t)N * 64;
    float*    ascore = skipb + (size_t)N * 64;
    unsigned* amax   = (unsigned*)(ascore + (size_t)E * 4);
    float*    denom  = (float*)(amax + (size_t)N * 4);
    float*    agg    = denom + (size_t)N * 4;

    // zero amax (key 0 < key(-inf)) + denom + agg in one contiguous memset
    hipMemsetAsync(amax, 0, (size_t)N * (4 + 4 + 64) * sizeof(float), stream);

    proj_wmma_kernel<<<(N + 15) / 16, 256, 0, stream>>>(feat, Wk, bk, Wsk, bsk, kdst, skipb, N);

    {
        int total = E * 4;
        score_max_kernel<<<(total + 255) / 256, 256, 0, stream>>>(q, kdst, src, dst, ascore, amax, E);
    }
    {
        int total = E * 64;
        agg_kernel<<<(total + 255) / 256, 256, 0, stream>>>(v, src, dst, ascore, amax, denom, agg, E);
    }
    finish_kernel<<<(N + 7) / 8, 256, 0, stream>>>(skipb, agg, denom, Wg, bg, gamma, beta, pa,
                                                   (float*)d_out, N);
}